// GATBackbone_58445914964271
// MI455X (gfx1250) — compile-verified
//
#include <hip/hip_runtime.h>
#include <hip/hip_bf16.h>

#define NN    20000      // N_NODES
#define NE    640000     // N_EDGES
#define HID   128
#define EDIM  32
#define HEADS 4
#define NEG_SLOPE 0.2f

typedef float v2f __attribute__((ext_vector_type(2)));
typedef float v8f __attribute__((ext_vector_type(8)));

#define WMMA_F32(a, b, c) \
    __builtin_amdgcn_wmma_f32_16x16x4_f32(false, (a), false, (b), (short)0, (c), false, false)

// ---- fused dual fp32-WMMA GEMM: xs = X@Ws, xd = X@Wd  (X: [M x 128]) ------
// Block = 256 threads = 8 waves; wave w owns the 16-col tile at 16*w.
// Each block covers 32 rows (two 16-row M-tiles): A fragments are reused for
// both weight matrices, B fragments are reused across both M-tiles, so each
// k-step issues 4 WMMAs for 6 loads.  M=20000 = 625 * 32 exactly.
__global__ __launch_bounds__(256)
void gemm128_dual_wmma(const float* __restrict__ X,
                       const float* __restrict__ Ws,
                       const float* __restrict__ Wd,
                       float* __restrict__ xs, float* __restrict__ xd) {
    const int wave = threadIdx.x >> 5;
    const int lane = threadIdx.x & 31;
    const int g    = lane >> 4;          // half-wave group (K select)
    const int mn   = lane & 15;          // row-in-tile (A) / col-in-tile (B,C)
    const int bm   = blockIdx.x * 32;
    const int bn   = wave * 16;

    const float* Arow0 = X + (bm + mn) * HID;
    const float* Arow1 = X + (bm + 16 + mn) * HID;
    const float* Bs    = Ws + bn + mn;
    const float* Bd    = Wd + bn + mn;

    v8f cs0 = {}, cs1 = {}, cd0 = {}, cd1 = {};
#pragma unroll
    for (int k0 = 0; k0 < HID; k0 += 4) {
        const int kk = k0 + 2 * g;
        v2f a0 = *(const v2f*)(Arow0 + kk);       // A[m][kk..kk+1]
        v2f a1 = *(const v2f*)(Arow1 + kk);
        v2f bs, bd;
        bs[0] = Bs[(kk + 0) * HID];               // B[kk  ][n]
        bs[1] = Bs[(kk + 1) * HID];               // B[kk+1][n]
        bd[0] = Bd[(kk + 0) * HID];
        bd[1] = Bd[(kk + 1) * HID];
        cs0 = WMMA_F32(a0, bs, cs0);
        cs1 = WMMA_F32(a1, bs, cs1);
        cd0 = WMMA_F32(a0, bd, cd0);
        cd1 = WMMA_F32(a1, bd, cd1);
    }
#pragma unroll
    for (int r = 0; r < 8; ++r) {
        const int row0 = bm + r + 8 * g;          // C layout per ISA
        xs[(row0     ) * HID + bn + mn] = cs0[r];
        xs[(row0 + 16) * HID + bn + mn] = cs1[r];
        xd[(row0     ) * HID + bn + mn] = cd0[r];
        xd[(row0 + 16) * HID + bn + mn] = cd1[r];
    }
}

// ---------------- monotone float<->uint encoding for atomic max ------------
__device__ __forceinline__ unsigned f2ord(float f) {
    unsigned u = __float_as_uint(f);
    return (u & 0x80000000u) ? ~u : (u | 0x80000000u);
}
__device__ __forceinline__ float ord2f(unsigned u) {
    return (u & 0x80000000u) ? __uint_as_float(u & 0x7fffffffu)
                             : __uint_as_float(~u);
}

// ---------------- per-layer init (out accumulator, denom, running max) -----
__global__ __launch_bounds__(256)
void init_layer(float* __restrict__ outb, float* __restrict__ denom,
                unsigned* __restrict__ mxu) {
    int t = blockIdx.x * 256 + threadIdx.x;
    outb[t] = 0.f;                       // grid covers NN*HID exactly
    if (t < NN * HEADS) { denom[t] = 0.f; mxu[t] = 0u; }  // 0 == -inf encoding
}

// ---------------- edge pass A: fused ee-GEMV + logits + atomic segment-max -
// One wave per edge. We (32x128) and att (128) staged in LDS. Lane j owns
// output channels 4j..4j+3 (all inside one head); per-head dot reduced over
// the 8 lanes of that head with xor-shuffles.
__global__ __launch_bounds__(256)
void edge_logits(const int* __restrict__ src, const int* __restrict__ dst,
                 const float* __restrict__ eattr,
                 const float* __restrict__ We,   // [32 x 128]
                 const float* __restrict__ att,  // [128] = [H][C] flattened
                 const float* __restrict__ xs, const float* __restrict__ xd,
                 float* __restrict__ logits, unsigned* __restrict__ mxu) {
    __shared__ float sW[EDIM * HID];
    __shared__ float sAtt[HID];
    for (int i = threadIdx.x; i < EDIM * HID; i += 256) sW[i] = We[i];
    if (threadIdx.x < HID) sAtt[threadIdx.x] = att[threadIdx.x];
    __syncthreads();

    const int wave = threadIdx.x >> 5;
    const int lane = threadIdx.x & 31;
    const int e = blockIdx.x * 8 + wave;       // NE divisible by 8
    const int s = src[e], d = dst[e];

    // edge-embedding GEMV: each lane loads one attr value, broadcasts it
    const float av = eattr[e * EDIM + lane];
    float e0 = 0.f, e1 = 0.f, e2 = 0.f, e3 = 0.f;
#pragma unroll
    for (int k = 0; k < EDIM; ++k) {
        float a = __shfl(av, k, 32);
        const float4 w = *(const float4*)&sW[k * HID + lane * 4];
        e0 += a * w.x; e1 += a * w.y; e2 += a * w.z; e3 += a * w.w;
    }
    const float4 vs = *(const float4*)&xs[s * HID + lane * 4];
    const float4 vd = *(const float4*)&xd[d * HID + lane * 4];
    const float4 at = *(const float4*)&sAtt[lane * 4];
    float m0 = vs.x + vd.x + e0; m0 = m0 > 0.f ? m0 : NEG_SLOPE * m0;
    float m1 = vs.y + vd.y + e1; m1 = m1 > 0.f ? m1 : NEG_SLOPE * m1;
    float m2 = vs.z + vd.z + e2; m2 = m2 > 0.f ? m2 : NEG_SLOPE * m2;
    float m3 = vs.w + vd.w + e3; m3 = m3 > 0.f ? m3 : NEG_SLOPE * m3;
    float p = m0 * at.x + m1 * at.y + m2 * at.z + m3 * at.w;
    // reduce across the 8 lanes belonging to this head
    p += __shfl_xor(p, 1, 32);
    p += __shfl_xor(p, 2, 32);
    p += __shfl_xor(p, 4, 32);
    if ((lane & 7) == 0) {
        const int h = lane >> 3;
        logits[e * HEADS + h] = p;
        atomicMax(&mxu[d * HEADS + h], f2ord(p));
    }
}

// ---------------- edge pass B: exp(logit - max) and atomic denom sum -------
__global__ __launch_bounds__(256)
void edge_expsum(const int* __restrict__ dst, float* __restrict__ logits,
                 const unsigned* __restrict__ mxu, float* __restrict__ denom) {
    int t = blockIdx.x * 256 + threadIdx.x;   // covers NE*HEADS exactly
    int e = t >> 2, h = t & 3;
    int d = dst[e];
    float ev = __expf(logits[t] - ord2f(mxu[d * HEADS + h]));
    logits[t] = ev;                            // reuse buffer for exp values
    atomicAdd(&denom[d * HEADS + h], ev);
}

// ---------------- edge pass C: alpha-weighted scatter of source messages ---
__global__ __launch_bounds__(256)
void edge_scatter(const int* __restrict__ src, const int* __restrict__ dst,
                  const float* __restrict__ evals,
                  const float* __restrict__ denom,
                  const float* __restrict__ xs, float* __restrict__ outb) {
    const int wave = threadIdx.x >> 5;
    const int lane = threadIdx.x & 31;
    const int e = blockIdx.x * 8 + wave;
    const int s = src[e], d = dst[e];
    const int h = lane >> 3;
    const float alpha = evals[e * HEADS + h] / denom[d * HEADS + h];
    const float4 vs = *(const float4*)&xs[s * HID + lane * 4];
    float* o = outb + d * HID + lane * 4;
    atomicAdd(o + 0, alpha * vs.x);
    atomicAdd(o + 1, alpha * vs.y);
    atomicAdd(o + 2, alpha * vs.z);
    atomicAdd(o + 3, alpha * vs.w);
}

// ---------------- node finalize: bias + ReLU -------------------------------
__global__ __launch_bounds__(256)
void node_finalize(const float* __restrict__ outb,
                   const float* __restrict__ bias, float* __restrict__ hout) {
    int t = blockIdx.x * 256 + threadIdx.x;   // covers NN*HID exactly
    float v = outb[t] + bias[t & (HID - 1)];
    hout[t] = v > 0.f ? v : 0.f;
}

extern "C" void kernel_launch(void* const* d_in, const int* in_sizes, int n_in,
                              void* d_out, int out_size, void* d_ws,
                              size_t ws_size, hipStream_t stream) {
    const float* x     = (const float*)d_in[0];
    const int*   ei    = (const int*)d_in[1];     // [2, NE]
    const float* eattr = (const float*)d_in[2];   // [NE, 32]
    // d_in[3] = batch (unused)
    const float* Wsrc  = (const float*)d_in[4];   // [3,128,128]
    const float* Wdst  = (const float*)d_in[5];   // [3,128,128]
    const float* Wedge = (const float*)d_in[6];   // [3,32,128]
    const float* att   = (const float*)d_in[7];   // [3,4,32]
    const float* bias  = (const float*)d_in[8];   // [3,128]
    const int* src = ei;
    const int* dst = ei + NE;

    float* ws = (float*)d_ws;
    float*    h      = ws;  ws += NN * HID;
    float*    xs     = ws;  ws += NN * HID;
    float*    xd     = ws;  ws += NN * HID;
    float*    outb   = ws;  ws += NN * HID;
    float*    logits = ws;  ws += NE * HEADS;
    float*    denom  = ws;  ws += NN * HEADS;
    unsigned* mxu    = (unsigned*)ws;             // NN * HEADS

    for (int l = 0; l < 3; ++l) {
        const float* hin = (l == 0) ? x : h;
        gemm128_dual_wmma<<<NN / 32, 256, 0, stream>>>(hin,
                                                       Wsrc + l * HID * HID,
                                                       Wdst + l * HID * HID,
                                                       xs, xd);
        init_layer<<<(NN * HID) / 256, 256, 0, stream>>>(outb, denom, mxu);
        edge_logits<<<NE / 8, 256, 0, stream>>>(src, dst, eattr,
                                                Wedge + l * EDIM * HID,
                                                att + l * HID, xs, xd,
                                                logits, mxu);
        edge_expsum<<<(NE * HEADS) / 256, 256, 0, stream>>>(dst, logits, mxu,
                                                            denom);
        edge_scatter<<<NE / 8, 256, 0, stream>>>(src, dst, logits, denom, xs,
                                                 outb);
        float* hout = (l == 2) ? (float*)d_out : h;
        node_finalize<<<(NN * HID) / 256, 256, 0, stream>>>(outb,
                                                            bias + l * HID,
                                                            hout);
    }
}